// KNN_45827301048337
// MI455X (gfx1250) — compile-verified
//
#include <hip/hip_runtime.h>

typedef __attribute__((ext_vector_type(2))) float v2f;
typedef __attribute__((ext_vector_type(4))) float v4f;
typedef __attribute__((ext_vector_type(8))) float v8f;

constexpr int BATCH  = 4;
constexpr int NQ     = 4096;   // queries per batch
constexpr int MS     = 8192;   // samples per batch
constexpr int DIM    = 128;
constexpr int KK     = 16;     // top-k
constexpr int WAVES  = 8;      // waves per block
constexpr int QPW    = 16;     // queries per wave (one WMMA M-tile)
constexpr int TILE   = 32;     // samples per N-tile (two WMMA N-tiles)
constexpr int NTILES = MS / TILE;
constexpr int KSTEPS = DIM / 4;
constexpr int LDSROW = 132;    // padded floats per sample row in LDS (bank-conflict free)
constexpr int DSTR   = 33;     // padded dist row
constexpr int CSTR   = 34;     // padded candidate row

// Phase 0: ||s||^2 per sample row into workspace.
__global__ __launch_bounds__(256) void KNN_s2_kernel(const float* __restrict__ s,
                                                     float* __restrict__ s2,
                                                     int total) {
  int i = blockIdx.x * 256 + threadIdx.x;
  if (i >= total) return;
  const v4f* p = (const v4f*)(s + (size_t)i * DIM);
  float acc = 0.f;
#pragma unroll
  for (int j = 0; j < DIM / 4; ++j) {
    v4f v = p[j];
    acc += v.x * v.x + v.y * v.y + v.z * v.z + v.w * v.w;
  }
  s2[i] = acc;
}

// Fused Q*S^T (fp32 WMMA) + async-LDS double-buffered staging + streaming top-16.
__global__ __launch_bounds__(256) void KNN_main_kernel(const float* __restrict__ q,
                                                       const float* __restrict__ s,
                                                       const float* __restrict__ s2,
                                                       int* __restrict__ out) {
  __shared__ float s_tile[2][TILE * LDSROW];         // double-buffered sample tiles
  __shared__ float s_s2t[2][TILE];                   // double-buffered ||s||^2 tiles
  __shared__ float s_dist[WAVES][QPW * DSTR];        // per-wave D-tile transpose scratch
  __shared__ float s_cd[WAVES][QPW][CSTR];           // candidate distances (2x16 per query)
  __shared__ int   s_ci[WAVES][QPW][CSTR];           // candidate indices

  const int tid  = threadIdx.x;
  const int wave = tid >> 5;
  const int lane = tid & 31;
  const int lq   = lane & 15;   // query/sample row within 16
  const int hi   = lane >> 4;   // half selector

  const int bid = blockIdx.x;
  const int b   = bid >> 5;                               // batch (32 blocks per batch)
  const int q0  = (bid & 31) * (WAVES * QPW) + wave * QPW;

  const float* sbatch  = s  + (size_t)b * MS * DIM;
  const float* s2batch = s2 + (size_t)b * MS;

  // Issue async (ASYNCcnt-tracked) fill of s_tile[bb]/s_s2t[bb] with tile at sm0.
  auto issue_tile = [&](int bb, int sm0) {
#pragma unroll
    for (int r = 0; r < 4; ++r) {
      int i   = tid + 256 * r;   // 1024 x 16B chunks, 32 per sample row
      int row = i >> 5;
      int c4  = i & 31;
      const float* gsrc = sbatch + (size_t)(sm0 + row) * DIM + c4 * 4;
      unsigned ldst = (unsigned)(size_t)&s_tile[bb][row * LDSROW + c4 * 4];
      asm volatile("global_load_async_to_lds_b128 %0, %1, off"
                   :: "v"(ldst), "v"(gsrc) : "memory");
    }
    if (tid < 8) {   // 32 floats of ||s||^2 = 8 x 16B
      const float* gs2 = s2batch + sm0 + tid * 4;
      unsigned ldst = (unsigned)(size_t)&s_s2t[bb][tid * 4];
      asm volatile("global_load_async_to_lds_b128 %0, %1, off"
                   :: "v"(ldst), "v"(gs2) : "memory");
    }
  };

  // A fragments in registers: lane L holds row (L%16), K-pair 4*ks + 2*hi .. +1
  const float* qbase = q + ((size_t)(b * NQ + q0 + lq)) * DIM + 2 * hi;
  v2f a[KSTEPS];
#pragma unroll
  for (int ks = 0; ks < KSTEPS; ++ks)
    a[ks] = *(const v2f*)(qbase + 4 * ks);

  // init candidate lists: lane owns query lq, half hi -> slots [16*hi, 16*hi+16)
#pragma unroll
  for (int i = 0; i < KK; ++i) {
    s_cd[wave][lq][16 * hi + i] = __builtin_inff();
    s_ci[wave][lq][16 * hi + i] = 0;
  }
  float curmax = __builtin_inff();
  int   maxpos = 16 * hi;

  issue_tile(0, 0);   // prologue: fill buffer 0 with tile 0

  for (int t = 0; t < NTILES; ++t) {
    const int cur = t & 1;
    const int sm0 = t * TILE;

    // My fills for buf[cur] complete; barrier => everyone's fills complete AND
    // everyone finished compute of tile t-1 (which read buf[cur^1]).
    asm volatile("s_wait_asynccnt 0" ::: "memory");
    __syncthreads();
    if (t + 1 < NTILES) issue_tile(cur ^ 1, sm0 + TILE);

    // 16x128 @ 128x32 via 32 K-steps of two V_WMMA_F32_16X16X4_F32.
    v8f c0 = {};
    v8f c1 = {};
    const float* bp0 = &s_tile[cur][lq * LDSROW + 2 * hi];
    const float* bp1 = bp0 + 16 * LDSROW;
#pragma unroll
    for (int ks = 0; ks < KSTEPS; ++ks) {
      v2f b0 = *(const v2f*)(bp0 + 4 * ks);
      v2f b1 = *(const v2f*)(bp1 + 4 * ks);
      c0 = __builtin_amdgcn_wmma_f32_16x16x4_f32(false, a[ks], false, b0,
                                                 (short)0, c0, false, false);
      c1 = __builtin_amdgcn_wmma_f32_16x16x4_f32(false, a[ks], false, b1,
                                                 (short)0, c1, false, false);
    }

    // Transpose D tiles into per-wave LDS: dist[q][s_local]
    {
      float* dw = s_dist[wave];
      const int qb = 8 * hi;
#pragma unroll
      for (int v = 0; v < 8; ++v) {
        dw[(qb + v) * DSTR + lq]      = c0[v];
        dw[(qb + v) * DSTR + 16 + lq] = c1[v];
      }
    }
    __builtin_amdgcn_wave_barrier();  // keep store->load program order

    // Streaming top-k: lane owns query lq over sample half hi of this tile.
    {
      const float* dr = &s_dist[wave][lq * DSTR + 16 * hi];
      float* cd = &s_cd[wave][lq][0];
      int*   ci = &s_ci[wave][lq][0];
      const int base = 16 * hi;
#pragma unroll 4
      for (int j = 0; j < 16; ++j) {
        float r = s_s2t[cur][base + j] - 2.0f * dr[j];  // ||s||^2 - 2 q.s (rank key)
        if (r < curmax) {
          cd[maxpos] = r;
          ci[maxpos] = sm0 + base + j;
          float m = cd[base];
          int   mp = base;
#pragma unroll
          for (int u = 1; u < 16; ++u) {
            float vv = cd[base + u];
            if (vv > m) { m = vv; mp = base + u; }
          }
          curmax = m;
          maxpos = mp;
        }
      }
    }
    __builtin_amdgcn_wave_barrier();
  }

  // Final merge: lanes 0..15 select 16 smallest of the 32 candidates, ascending order.
  __builtin_amdgcn_wave_barrier();
  if (lane < 16) {
    float* cd = &s_cd[wave][lane][0];
    int*   ci = &s_ci[wave][lane][0];
    int* outp = out + ((size_t)(b * NQ + q0 + lane)) * KK;
    for (int sel = 0; sel < KK; ++sel) {
      float m = cd[0];
      int   mp = 0;
#pragma unroll
      for (int u = 1; u < 32; ++u) {
        float vv = cd[u];
        if (vv < m) { m = vv; mp = u; }
      }
      outp[sel] = ci[mp];
      cd[mp] = __builtin_inff();
    }
  }
}

extern "C" void kernel_launch(void* const* d_in, const int* in_sizes, int n_in,
                              void* d_out, int out_size, void* d_ws, size_t ws_size,
                              hipStream_t stream) {
  const float* qpts = (const float*)d_in[0];
  const float* spts = (const float*)d_in[1];
  float* s2  = (float*)d_ws;            // BATCH*MS floats = 128 KB
  int*   out = (int*)d_out;             // [B, N, 16] int32

  int total = BATCH * MS;
  KNN_s2_kernel<<<(total + 255) / 256, 256, 0, stream>>>(spts, s2, total);

  int blocks = BATCH * (NQ / (WAVES * QPW));  // 4 * 32 = 128
  KNN_main_kernel<<<blocks, 256, 0, stream>>>(qpts, spts, s2, out);
}